// BotGCN_6957847019951
// MI455X (gfx1250) — compile-verified
//
#include <hip/hip_runtime.h>

typedef __attribute__((ext_vector_type(2))) float v2f;
typedef __attribute__((ext_vector_type(8))) float v8f;

#define FEAT_IN  128
#define FEAT_HID 64
#define FEAT_OUT 2

// ---------------- degree / normalization ----------------------------------

__global__ void k_init_deg(float* __restrict__ deg, int N) {
    int i = blockIdx.x * blockDim.x + threadIdx.x;
    if (i < N) deg[i] = 1.0f;                       // self-loop counts once
}

__global__ void k_count_deg(const long long* __restrict__ dst,
                            float* __restrict__ deg, int E) {
    int e = blockIdx.x * blockDim.x + threadIdx.x;
    if (e < E) atomicAdd(&deg[(int)dst[e]], 1.0f);
}

__global__ void k_finalize_dinv(float* __restrict__ deg, int N) {
    int i = blockIdx.x * blockDim.x + threadIdx.x;
    if (i < N) deg[i] = rsqrtf(deg[i]);             // deg >= 1 always
}

// ---------------- layer-1 GEMM: h1 = x @ W1 via f32 WMMA -------------------
// One wave computes one 16x16 tile of h1. K=128 swept in steps of 4 with
// V_WMMA_F32_16X16X4_F32 (full fp32 precision, matches reference).
// Grid: (N/16)*(64/16) tiles, 8 waves (tiles) per 256-thread block.

__global__ __launch_bounds__(256) void k_gemm1_wmma(
        const float* __restrict__ X,   // [N,128]
        const float* __restrict__ W,   // [128,64]
        float* __restrict__ H,         // [N,64]
        int tilesTotal) {
    const int wave = threadIdx.x >> 5;
    const int tile = blockIdx.x * 8 + wave;
    if (tile >= tilesTotal) return;            // wave-uniform guard: EXEC stays full

    const int tn   = tile & 3;                 // 64/16 = 4 col tiles
    const int tm   = tile >> 2;                // row tile
    const int lane = threadIdx.x & 31;
    const int half = lane >> 4;                // 0: K lo pair, 1: K hi pair
    const int l    = lane & 15;

    const float* Arow = X + (size_t)(tm * 16 + l) * FEAT_IN + 2 * half;
    const float* Bcol = W + (size_t)(2 * half) * FEAT_HID + tn * 16 + l;

    v8f acc = {};
#pragma unroll
    for (int k0 = 0; k0 < FEAT_IN; k0 += 4) {
        v2f a, b;
        // A 16x4: lane half selects K pair {k0,k0+1} or {k0+2,k0+3}, row = l
        a.x = Arow[k0 + 0];
        a.y = Arow[k0 + 1];
        // B 4x16: same K pair striped across the 2 VGPRs, col = l
        b.x = Bcol[(size_t)k0 * FEAT_HID];
        b.y = Bcol[(size_t)(k0 + 1) * FEAT_HID];
        acc = __builtin_amdgcn_wmma_f32_16x16x4_f32(
                  false, a, false, b, (short)0, acc, false, false);
    }

    // C/D layout: VGPR r -> row r (lanes 0-15) / row 8+r (lanes 16-31), col l
    float* out = H + (size_t)(tm * 16 + half * 8) * FEAT_HID + tn * 16 + l;
#pragma unroll
    for (int r = 0; r < 8; ++r) out[(size_t)r * FEAT_HID] = acc[r];
}

// ---------------- layer-1 aggregation --------------------------------------

__global__ void k_selfloop1(const float* __restrict__ h1,
                            const float* __restrict__ dinv,
                            float* __restrict__ agg1, int total /* N*64 */) {
    int g = blockIdx.x * blockDim.x + threadIdx.x;
    if (g < total) {
        int i = g >> 6;
        float di = dinv[i];
        agg1[g] = h1[g] * di * di;              // self-loop norm = dinv^2
    }
}

__global__ void k_edge_agg1(const long long* __restrict__ src,
                            const long long* __restrict__ dst,
                            const float* __restrict__ h1,
                            const float* __restrict__ dinv,
                            float* __restrict__ agg1, int E) {
    int g = blockIdx.x * blockDim.x + threadIdx.x;   // E*64 = 64M < 2^31
    if (g < E * FEAT_HID) {
        int e = g >> 6;
        int f = g & 63;
        int s = (int)src[e];
        int d = (int)dst[e];
        float norm = dinv[s] * dinv[d];
        atomicAdd(&agg1[(size_t)d * FEAT_HID + f],
                  h1[(size_t)s * FEAT_HID + f] * norm);
    }
}

// ---------------- layer-2: relu(agg1+b1) @ W2 ------------------------------
// N-dim is 2; a scalar dual-accumulator dot per node beats a 16-wide WMMA
// tile that would be 87% padding. Bias+ReLU fused into the read.

__global__ void k_gemm2(const float* __restrict__ agg1,
                        const float* __restrict__ b1,
                        const float* __restrict__ W2,  // [64,2]
                        float* __restrict__ h2,        // [N,2]
                        int N) {
    int i = blockIdx.x * blockDim.x + threadIdx.x;
    if (i < N) {
        const float* row = agg1 + (size_t)i * FEAT_HID;
        float acc0 = 0.f, acc1 = 0.f;
#pragma unroll
        for (int k = 0; k < FEAT_HID; ++k) {
            float a = row[k] + b1[k];
            a = a > 0.f ? a : 0.f;              // ReLU
            acc0 = fmaf(a, W2[k * 2 + 0], acc0);
            acc1 = fmaf(a, W2[k * 2 + 1], acc1);
        }
        h2[(size_t)i * 2 + 0] = acc0;
        h2[(size_t)i * 2 + 1] = acc1;
    }
}

// ---------------- layer-2 aggregation --------------------------------------

__global__ void k_selfloop2(const float* __restrict__ h2,
                            const float* __restrict__ dinv,
                            const float* __restrict__ b2,
                            float* __restrict__ out, int total /* N*2 */) {
    int g = blockIdx.x * blockDim.x + threadIdx.x;
    if (g < total) {
        int i = g >> 1;
        int c = g & 1;
        float di = dinv[i];
        out[g] = h2[g] * di * di + b2[c];
    }
}

__global__ void k_edge_agg2(const long long* __restrict__ src,
                            const long long* __restrict__ dst,
                            const float* __restrict__ h2,
                            const float* __restrict__ dinv,
                            float* __restrict__ out, int E) {
    int g = blockIdx.x * blockDim.x + threadIdx.x;   // E*2
    if (g < E * FEAT_OUT) {
        int e = g >> 1;
        int c = g & 1;
        int s = (int)src[e];
        int d = (int)dst[e];
        atomicAdd(&out[(size_t)d * FEAT_OUT + c],
                  h2[(size_t)s * FEAT_OUT + c] * dinv[s] * dinv[d]);
    }
}

// ---------------- driver ----------------------------------------------------

extern "C" void kernel_launch(void* const* d_in, const int* in_sizes, int n_in,
                              void* d_out, int out_size, void* d_ws, size_t ws_size,
                              hipStream_t stream) {
    const float*     x   = (const float*)d_in[0];
    const float*     W1  = (const float*)d_in[1];
    const float*     b1  = (const float*)d_in[2];
    const float*     W2  = (const float*)d_in[3];
    const float*     b2  = (const float*)d_in[4];
    const long long* ei  = (const long long*)d_in[5];   // int64 [2, E]

    const int N = in_sizes[0] / FEAT_IN;     // 100000
    const int E = in_sizes[5] / 2;           // 1000000
    const long long* srcIdx = ei;
    const long long* dstIdx = ei + E;
    float* out = (float*)d_out;

    // workspace: dinv[N] | h1[N*64] | agg1[N*64] | h2[N*2]  (~52.4 MB)
    float* dinv = (float*)d_ws;
    float* h1   = dinv + N;
    float* agg1 = h1 + (size_t)N * FEAT_HID;
    float* h2   = agg1 + (size_t)N * FEAT_HID;

    const int B = 256;

    // 1) symmetric normalization: dinv = rsqrt(in-degree incl. self-loop)
    k_init_deg<<<(N + B - 1) / B, B, 0, stream>>>(dinv, N);
    k_count_deg<<<(E + B - 1) / B, B, 0, stream>>>(dstIdx, dinv, E);
    k_finalize_dinv<<<(N + B - 1) / B, B, 0, stream>>>(dinv, N);

    // 2) h1 = x @ W1 (fp32 WMMA)
    const int tiles = (N / 16) * (FEAT_HID / 16);      // 25000
    k_gemm1_wmma<<<(tiles + 7) / 8, 256, 0, stream>>>(x, W1, h1, tiles);

    // 3) layer-1 aggregation (self-loop init + edge scatter-add)
    k_selfloop1<<<((size_t)N * FEAT_HID + B - 1) / B, B, 0, stream>>>(
        h1, dinv, agg1, N * FEAT_HID);
    k_edge_agg1<<<(int)(((long long)E * FEAT_HID + B - 1) / B), B, 0, stream>>>(
        srcIdx, dstIdx, h1, dinv, agg1, E);

    // 4) h2 = relu(agg1 + b1) @ W2
    k_gemm2<<<(N + B - 1) / B, B, 0, stream>>>(agg1, b1, W2, h2, N);

    // 5) layer-2 aggregation -> logits
    k_selfloop2<<<(N * FEAT_OUT + B - 1) / B, B, 0, stream>>>(
        h2, dinv, b2, out, N * FEAT_OUT);
    k_edge_agg2<<<(E * FEAT_OUT + B - 1) / B, B, 0, stream>>>(
        srcIdx, dstIdx, h2, dinv, out, E);
}